// SemiHardTripletMiner_14851996910158
// MI455X (gfx1250) — compile-verified
//
#include <hip/hip_runtime.h>
#include <hip/hip_bf16.h>

// Problem constants (reference: N=512, D=128, NUM_CLASSES=64, MARGIN=1.0)
#define N 512
#define D 128
#define MARGIN 1.0f

typedef __attribute__((ext_vector_type(2))) float v2f;
typedef __attribute__((ext_vector_type(8))) float v8f;

// ---------------------------------------------------------------------------
// Kernel 0: row squared-norms. One wave per row, each lane loads float4.
// ---------------------------------------------------------------------------
__global__ __launch_bounds__(256) void triplet_norms_kernel(
    const float* __restrict__ X, float* __restrict__ nrm) {
  const int wave = threadIdx.x >> 5;
  const int lane = threadIdx.x & 31;
  const int row = blockIdx.x * 8 + wave;          // 64 blocks * 8 waves = 512 rows
  const float4* p = (const float4*)(X + row * D); // D=128 -> 32 float4 per row
  float4 v = p[lane];
  float s = v.x * v.x + v.y * v.y + v.z * v.z + v.w * v.w;
#pragma unroll
  for (int off = 16; off > 0; off >>= 1) s += __shfl_xor(s, off, 32);
  if (lane == 0) nrm[row] = s;
}

// ---------------------------------------------------------------------------
// Kernel 1: pairwise distances via V_WMMA_F32_16X16X4_F32.
// One wave (32 threads) per 16x16 tile of G = X * X^T; 32 k-steps of K=4.
// A-matrix 16x4 f32 layout: VGPR r, lane L -> A[M = L&15][K = r + 2*(L>>4)]
// B-matrix 4x16 f32 layout: VGPR r, lane L -> B[K = r + 2*(L>>4)][N = L&15]
// For the Gram matrix B[k][n] = X[n][k], so both loads are row-major on X.
// C/D 16x16 f32 layout: VGPR r, lane L -> M = r + 8*(L>>4), N = L&15.
// ---------------------------------------------------------------------------
__global__ __launch_bounds__(32) void triplet_pd_wmma_kernel(
    const float* __restrict__ X, const float* __restrict__ nrm,
    float* __restrict__ pd) {
  const int tn = blockIdx.x;            // column tile (N/16 = 32)
  const int tm = blockIdx.y;            // row tile
  const int lane = threadIdx.x;         // 0..31, wave32
  const int half = lane >> 4;           // 0 or 1
  const int l15 = lane & 15;

  const float* arow = X + (tm * 16 + l15) * D + half * 2;  // A rows
  const float* brow = X + (tn * 16 + l15) * D + half * 2;  // B "rows" (X row-major)

  v8f c = {};
#pragma unroll 4
  for (int k0 = 0; k0 < D; k0 += 4) {
    v2f a, b;
    a.x = arow[k0];
    a.y = arow[k0 + 1];
    b.x = brow[k0];
    b.y = brow[k0 + 1];
    // D = A(16x4) * B(4x16) + C, full fp32
    c = __builtin_amdgcn_wmma_f32_16x16x4_f32(
        /*neg_a=*/false, a, /*neg_b=*/false, b,
        /*c_mod=*/(short)0, c, /*reuse_a=*/false, /*reuse_b=*/false);
  }

  const int j = tn * 16 + l15;
  const float nj = nrm[j];
#pragma unroll
  for (int r = 0; r < 8; ++r) {
    const int i = tm * 16 + r + 8 * half;
    const float g = c[r];
    float d2 = nrm[i] + nj - 2.0f * g;
    float dist = sqrtf(fmaxf(d2, 0.0f));
    pd[i * N + j] = (i == j) ? 0.0f : dist;
  }
}

// ---------------------------------------------------------------------------
// Kernel 2: semi-hard mining. One block (256 threads) per anchor `a`.
// Deterministic: positive list built serially by thread 0; per-anchor partial
// sums written out, no float atomics anywhere.
// ---------------------------------------------------------------------------
__global__ __launch_bounds__(256) void triplet_mine_kernel(
    const float* __restrict__ pd, const int* __restrict__ labels,
    float* __restrict__ loss_out, float* __restrict__ cnt_out) {
  __shared__ float row[N];
  __shared__ unsigned char notEq[N];
  __shared__ float wred[8];
  __shared__ int wany[8];
  __shared__ float s_mx, s_mn, s_ni, s_loss;
  __shared__ int posList[N];
  __shared__ int s_npos;

  const int a = blockIdx.x;
  const int t = threadIdx.x;
  const int wave = t >> 5;
  const int lane = t & 31;
  const int la = labels[a];

  for (int j = t; j < N; j += 256) {
    row[j] = pd[a * N + j];
    notEq[j] = (labels[j] != la) ? 1 : 0;
  }
  __syncthreads();

  // ---- row max / min (axis_max, axis_min) ----
  float mx = -3.4e38f, mn = 3.4e38f;
  for (int j = t; j < N; j += 256) {
    mx = fmaxf(mx, row[j]);
    mn = fminf(mn, row[j]);
  }
#pragma unroll
  for (int off = 16; off > 0; off >>= 1) {
    mx = fmaxf(mx, __shfl_xor(mx, off, 32));
    mn = fminf(mn, __shfl_xor(mn, off, 32));
  }
  if (lane == 0) wred[wave] = mx;
  __syncthreads();
  if (t == 0) {
    float m = wred[0];
    for (int w = 1; w < 8; ++w) m = fmaxf(m, wred[w]);
    s_mx = m;
  }
  __syncthreads();
  if (lane == 0) wred[wave] = mn;
  __syncthreads();
  if (t == 0) {
    float m = wred[0];
    for (int w = 1; w < 8; ++w) m = fminf(m, wred[w]);
    s_mn = m;
  }
  __syncthreads();
  const float rmax = s_mx;
  const float rmin = s_mn;

  // ---- negatives_inside = max(0, max over notEq of row[k]-rmin) + rmin ----
  float mi = 0.0f;  // cond==0 entries contribute 0 to jnp.max
  for (int k = t; k < N; k += 256)
    if (notEq[k]) mi = fmaxf(mi, row[k] - rmin);
#pragma unroll
  for (int off = 16; off > 0; off >>= 1) mi = fmaxf(mi, __shfl_xor(mi, off, 32));
  if (lane == 0) wred[wave] = mi;
  __syncthreads();
  if (t == 0) {
    float m = wred[0];
    for (int w = 1; w < 8; ++w) m = fmaxf(m, wred[w]);
    s_ni = m + rmin;
    s_loss = 0.0f;
    // deterministic positive list (same label, b != a)
    int np = 0;
    for (int b = 0; b < N; ++b)
      if (!notEq[b] && b != a) posList[np++] = b;
    s_npos = np;
  }
  __syncthreads();
  const float ninside = s_ni;
  const int npos = s_npos;

  // ---- per-positive: masked min over candidates (all 256 threads cooperate)
  for (int p = 0; p < npos; ++p) {
    const int b = posList[p];
    const float pab = row[b];
    float mo = 0.0f;  // min(shifted*cond): cond==0 entries contribute 0
    int any = 0;
    for (int k = t; k < N; k += 256) {
      if (notEq[k] && row[k] > pab) {
        any = 1;
        mo = fminf(mo, row[k] - rmax);  // shifted = pd - rowmax <= 0
      }
    }
#pragma unroll
    for (int off = 16; off > 0; off >>= 1) {
      mo = fminf(mo, __shfl_xor(mo, off, 32));
      any |= __shfl_xor(any, off, 32);
    }
    if (lane == 0) { wred[wave] = mo; wany[wave] = any; }
    __syncthreads();
    if (t == 0) {
      float M = wred[0];
      int A = wany[0];
      for (int w = 1; w < 8; ++w) { M = fminf(M, wred[w]); A |= wany[w]; }
      const float shn = A ? (M + rmax) : ninside;
      const float lv = MARGIN + pab - shn;
      s_loss += fmaxf(lv, 0.0f);
    }
    __syncthreads();
  }

  if (t == 0) {
    loss_out[a] = s_loss;
    cnt_out[a] = (float)npos;
  }
}

// ---------------------------------------------------------------------------
// Kernel 3: deterministic final reduction -> scalar loss.
// ---------------------------------------------------------------------------
__global__ __launch_bounds__(256) void triplet_finalize_kernel(
    const float* __restrict__ loss, const float* __restrict__ cnt,
    float* __restrict__ out) {
  __shared__ float sl[256], sc[256];
  const int t = threadIdx.x;
  sl[t] = loss[t] + loss[t + 256];
  sc[t] = cnt[t] + cnt[t + 256];
  __syncthreads();
  for (int s = 128; s > 0; s >>= 1) {
    if (t < s) { sl[t] += sl[t + s]; sc[t] += sc[t + s]; }
    __syncthreads();
  }
  if (t == 0) out[0] = sl[0] / sc[0];
}

// ---------------------------------------------------------------------------
extern "C" void kernel_launch(void* const* d_in, const int* in_sizes, int n_in,
                              void* d_out, int out_size, void* d_ws, size_t ws_size,
                              hipStream_t stream) {
  const float* X = (const float*)d_in[0];     // embeddings [512,128] f32
  const int* labels = (const int*)d_in[1];    // labels [512] i32
  float* out = (float*)d_out;                 // scalar f32

  char* ws = (char*)d_ws;
  float* nrm = (float*)ws;                              // 512 f32
  float* pd = (float*)(ws + 2048);                      // 512*512 f32 = 1 MB
  float* loss_part = (float*)(ws + 2048 + N * N * 4);   // 512 f32
  float* cnt_part = (float*)(ws + 2048 + N * N * 4 + N * 4);  // 512 f32

  triplet_norms_kernel<<<64, 256, 0, stream>>>(X, nrm);
  triplet_pd_wmma_kernel<<<dim3(N / 16, N / 16, 1), 32, 0, stream>>>(X, nrm, pd);
  triplet_mine_kernel<<<N, 256, 0, stream>>>(pd, labels, loss_part, cnt_part);
  triplet_finalize_kernel<<<1, 256, 0, stream>>>(loss_part, cnt_part, out);
}